// CustomFFTConv2d_17695265259673
// MI455X (gfx1250) — compile-verified
//
#include <hip/hip_runtime.h>

// CDNA5 / gfx1250 implicit-GEMM implementation of the "custom FFT conv".
//
// Math (derived from the reference): the custom ifft2 flips the h axis of the
// true spatial result S (linear 2D conv of padded x with padded 3x3 weights):
//   out[b,o,i,j] = bias[o] + sum_{c,p,q} W[o,c,p,q] * x[b,c, 96-i-p, 32+j-q]
// with x treated as zero outside [0,63]^2.  Non-trivial only for
// i in [31,63], j in [0,33]  (33 x 34 = 1122 pixels per batch);
// everything else is bias[o].
//
// GEMM view: M = OC = 64, N = 1122 pixels (tiled by 16), K = 288.
// K-ordering k = t*32 + c (t = p*3+q tap, c = channel):
//   * (c,t) decode is shift/and, constant after full unroll
//   * all load addresses = per-lane base + 24-bit immediate offsets
//   * boundary mask depends only on t -> one exec-masked load block per tap
//
// Parallelism: 4 waves per block split the K dimension (8 channels each) and
// reduce partial accumulators through LDS — same total loads per block as a
// single-wave version, but 4x the wave count (4544 waves) for latency hiding.
//
// Kernel 1: stream bias into the full [16,64,64,64] output (float4 stores).
// Kernel 2: fp32 implicit GEMM, V_WMMA_F32_16X16X4_F32, K-split + LDS reduce.

typedef float v2f __attribute__((ext_vector_type(2)));
typedef float v8f __attribute__((ext_vector_type(8)));

#define B_DIM   16
#define IC      32
#define OC      64
#define HW      64
#define KDIM    288            // IC * 3 * 3
#define ACT_I0  31
#define ACT_NJ  34
#define NPIX    1122           // 33 * 34 active pixels per batch
#define NTILES  71             // ceil(1122 / 16)
#define LSTRIDE 36             // padded lane stride (floats) in LDS reduce

// ---------------------------------------------------------------------------
// Kernel 1: out[b,o,i,j] = bias[o] for the entire output (float4 vectorized).
// ---------------------------------------------------------------------------
__global__ __launch_bounds__(256) void bias_fill_kernel(
    const float* __restrict__ bias, float* __restrict__ out)
{
    int idx4 = blockIdx.x * 256 + threadIdx.x;      // float4 index
    int o = (idx4 >> 10) & (OC - 1);                // (idx4*4 >> 12) & 63
    float bv = bias[o];
    float4 v = make_float4(bv, bv, bv, bv);
    reinterpret_cast<float4*>(out)[idx4] = v;
}

// ---------------------------------------------------------------------------
// Kernel 2: 4 waves per block; N-tile = 16 pixels (blockIdx.x), batch =
// blockIdx.y.  Wave ws owns channels [8*ws, 8*ws+8) (K slice of 72) and all
// four 16-row M-tiles, so each B fragment is reused by 4 back-to-back WMMAs.
// Partial accumulators are summed through LDS; wave 0 writes the output.
//
// V_WMMA_F32_16X16X4_F32 fragment layouts (ISA 7.12.2):
//   A (16x4):  lanes 0-15 -> M=lane, {v0,v1}={K+0,K+1}; lanes 16-31 -> {K+2,K+3}
//   B (4x16):  lanes 0-15 -> N=lane, {v0,v1}={K+0,K+1}; lanes 16-31 -> {K+2,K+3}
//   C/D:       vgpr r: lanes 0-15 -> (M=r, N=lane), lanes 16-31 -> (M=r+8)
// ---------------------------------------------------------------------------
__global__ __launch_bounds__(128) void conv_wmma_kernel(
    const float* __restrict__ x,      // [16][32][64][64]
    const float* __restrict__ w,      // [64][32][3][3] = [64][288]
    const float* __restrict__ bias,   // [64]
    float* __restrict__ out)          // [16][64][64][64]
{
    __shared__ float red[3 * 32 * LSTRIDE];          // 13.8 KB partial exchange

    const int tid      = threadIdx.x;
    const int wave     = tid >> 5;           // 0..3 : K-slice owner
    const int lane     = tid & 31;
    const int laneN    = lane & 15;          // column / A-row within tile
    const int half     = lane >> 4;          // 0: K+{0,1}, 1: K+{2,3}
    const int kofs     = half * 2;
    const int cw       = wave * 8;           // first channel of this wave's slice

    const int b        = blockIdx.y;
    const int tileBase = blockIdx.x * 16;

    const int pix   = tileBase + laneN;
    const int valid = pix < NPIX;
    const int i     = ACT_I0 + pix / ACT_NJ;   // output row  (31..63)
    const int j     = pix % ACT_NJ;            // output col  (0..33)

    // Per-lane base pointers; everything else is immediate offsets.
    const float* __restrict__ xbase =
        x + (size_t)b * (IC * HW * HW) + (size_t)(cw + kofs) * (HW * HW)
          + (96 - i) * HW + (32 + j);
    const float* __restrict__ wbase = w + laneN * KDIM + (cw + kofs) * 9;

    v8f acc0 = {0.f, 0.f, 0.f, 0.f, 0.f, 0.f, 0.f, 0.f};
    v8f acc1 = {0.f, 0.f, 0.f, 0.f, 0.f, 0.f, 0.f, 0.f};
    v8f acc2 = {0.f, 0.f, 0.f, 0.f, 0.f, 0.f, 0.f, 0.f};
    v8f acc3 = {0.f, 0.f, 0.f, 0.f, 0.f, 0.f, 0.f, 0.f};

#pragma unroll
    for (int t = 0; t < 9; ++t) {            // tap index: p = t/3, q = t%3
        const int p = t / 3;
        const int q = t - 3 * p;

        // x row = 96-i-p in [0,63]  <=>  i >= 33-p
        // x col = 32+j-q in [0,63]  <=>  j <= 31+q   (lower bounds always ok)
        const int ok = valid & (i >= 33 - p) & (j <= 31 + q);

        // Gather this tap's x values (2 c-groups x 2 channels) in one
        // exec-masked region; masked lanes keep their zero init.
        float bv[4];
#pragma unroll
        for (int u = 0; u < 4; ++u) bv[u] = 0.f;
        if (ok) {
#pragma unroll
            for (int u = 0; u < 2; ++u) {
                bv[2 * u]     = xbase[(u * 4 + 0) * (HW * HW) - p * HW - q];
                bv[2 * u + 1] = xbase[(u * 4 + 1) * (HW * HW) - p * HW - q];
            }
        }

        // 2 K-steps for this tap: k = t*32 + cw + 4u + {kofs, kofs+1}
#pragma unroll
        for (int u = 0; u < 2; ++u) {
            const int c = u * 4;             // cw+kofs already folded into bases
            v2f bf = {bv[2 * u], bv[2 * u + 1]};

            // A element for k = t*32+c' is Wflat[row*288 + c'*9 + t]
            v2f a0 = {wbase[0 * 16 * KDIM + c * 9 + t],
                      wbase[0 * 16 * KDIM + (c + 1) * 9 + t]};
            v2f a1 = {wbase[1 * 16 * KDIM + c * 9 + t],
                      wbase[1 * 16 * KDIM + (c + 1) * 9 + t]};
            v2f a2 = {wbase[2 * 16 * KDIM + c * 9 + t],
                      wbase[2 * 16 * KDIM + (c + 1) * 9 + t]};
            v2f a3 = {wbase[3 * 16 * KDIM + c * 9 + t],
                      wbase[3 * 16 * KDIM + (c + 1) * 9 + t]};

            acc0 = __builtin_amdgcn_wmma_f32_16x16x4_f32(false, a0, false, bf,
                                                         (short)0, acc0, false, false);
            acc1 = __builtin_amdgcn_wmma_f32_16x16x4_f32(false, a1, false, bf,
                                                         (short)0, acc1, false, false);
            acc2 = __builtin_amdgcn_wmma_f32_16x16x4_f32(false, a2, false, bf,
                                                         (short)0, acc2, false, false);
            acc3 = __builtin_amdgcn_wmma_f32_16x16x4_f32(false, a3, false, bf,
                                                         (short)0, acc3, false, false);
        }
    }

    // ---- Cross-wave K reduction through LDS (b128 traffic, padded stride)
    if (wave != 0) {
        float buf[32];
#pragma unroll
        for (int r = 0; r < 8; ++r) {
            buf[r]      = acc0[r];
            buf[8 + r]  = acc1[r];
            buf[16 + r] = acc2[r];
            buf[24 + r] = acc3[r];
        }
        float4* dst = (float4*)&red[(wave - 1) * (32 * LSTRIDE) + lane * LSTRIDE];
#pragma unroll
        for (int v = 0; v < 8; ++v) dst[v] = ((const float4*)buf)[v];
    }
    __syncthreads();

    if (wave == 0) {
#pragma unroll
        for (int s = 0; s < 3; ++s) {
            const float4* src =
                (const float4*)&red[s * (32 * LSTRIDE) + lane * LSTRIDE];
            float buf[32];
#pragma unroll
            for (int v = 0; v < 8; ++v) ((float4*)buf)[v] = src[v];
#pragma unroll
            for (int r = 0; r < 8; ++r) {
                acc0[r] += buf[r];
                acc1[r] += buf[8 + r];
                acc2[r] += buf[16 + r];
                acc3[r] += buf[24 + r];
            }
        }

        // ---- Epilogue: D layout -> out[b, o, i, j] = acc + bias[o]
        if (valid) {
            const size_t plane = (size_t)HW * HW;
            float* __restrict__ ob = out + (((size_t)b * OC) * HW + i) * HW + j;
#pragma unroll
            for (int r = 0; r < 8; ++r) {
                int m = r + half * 8;        // row within 16-row M-tile
                ob[(0 * 16 + m) * plane] = acc0[r] + bias[0 * 16 + m];
                ob[(1 * 16 + m) * plane] = acc1[r] + bias[1 * 16 + m];
                ob[(2 * 16 + m) * plane] = acc2[r] + bias[2 * 16 + m];
                ob[(3 * 16 + m) * plane] = acc3[r] + bias[3 * 16 + m];
            }
        }
    }
}

extern "C" void kernel_launch(void* const* d_in, const int* in_sizes, int n_in,
                              void* d_out, int out_size, void* d_ws, size_t ws_size,
                              hipStream_t stream)
{
    const float* x    = (const float*)d_in[0];   // 16*32*64*64
    const float* w    = (const float*)d_in[1];   // 64*32*3*3
    const float* bias = (const float*)d_in[2];   // 64
    float* out        = (float*)d_out;           // 16*64*64*64

    // 1) bias broadcast over the full output (float4 stores)
    const int n4 = (B_DIM * OC * HW * HW) / 4;   // 1,048,576 float4s
    bias_fill_kernel<<<n4 / 256, 256, 0, stream>>>(bias, out);

    // 2) fp32 WMMA implicit GEMM over the active 33x34 region
    //    (4 waves/block K-split, LDS reduction)
    conv_wmma_kernel<<<dim3(NTILES, B_DIM), 128, 0, stream>>>(x, w, bias, out);
}